// CertBP_22445499089474
// MI455X (gfx1250) — compile-verified
//
#include <hip/hip_runtime.h>
#include <hip/hip_bf16.h>
#include <math.h>

// ---------------- problem constants (match reference) ----------------
#define IN_DIM   256
#define HID      128
#define NC       10
#define EMIN     258     // 2*HID + 2
#define EH       64      // EDGE_HID
#define EPS_     1e-12f
#define W_MAX_   0.8f
#define ALPHA_MAX_ 1.5f
#define EXP_CLIP_ 20.0f
#define UNARY_TEMP_ 1.5f
#define T_ITERS  10
#define ETA_     0.2f

typedef __attribute__((ext_vector_type(2))) float v2f;
typedef __attribute__((ext_vector_type(8))) float v8f;

// ---------------- generic zero ----------------
__global__ void k_zero(float* __restrict__ p, int n) {
    int i = blockIdx.x * blockDim.x + threadIdx.x;
    if (i < n) p[i] = 0.0f;
}

// ---------------- encoder GEMM: h = relu(x @ W1 + b1), WMMA f32 16x16x4 ----------------
// One wave computes a 16-row x 128-col tile of h. 8 accumulators (v8f) over 64 K-steps.
__global__ void __launch_bounds__(128)
k_enc_gemm(const float* __restrict__ x, const float* __restrict__ w1,
           const float* __restrict__ b1, float* __restrict__ h,
           int ntiles, int nnodes) {
    int wave = threadIdx.x >> 5;
    int lane = threadIdx.x & 31;
    int tile = blockIdx.x * 4 + wave;
    if (tile >= ntiles) return;               // uniform per wave -> EXEC all-ones for WMMA
    int half = lane >> 4;                     // K-pair selector
    int l    = lane & 15;

    int arow = tile * 16 + l;
    if (arow >= nnodes) arow = nnodes - 1;    // clamp (N is a multiple of 16 here anyway)
    const float* xr = x + (size_t)arow * IN_DIM;

    v8f acc[8];
    v8f zero = 0.0f;
#pragma unroll
    for (int n = 0; n < 8; ++n) acc[n] = zero;

    for (int k = 0; k < IN_DIM; k += 4) {
        int ko = k + 2 * half;                // this lane-half supplies K=ko, ko+1
        v2f a;
        a.x = xr[ko];
        a.y = xr[ko + 1];
        const float* wr0 = w1 + (size_t)ko * HID;
        const float* wr1 = wr0 + HID;
#pragma unroll
        for (int n = 0; n < 8; ++n) {
            v2f b;
            b.x = wr0[n * 16 + l];
            b.y = wr1[n * 16 + l];
            acc[n] = __builtin_amdgcn_wmma_f32_16x16x4_f32(
                false, a, false, b, (short)0, acc[n], false, false);
        }
    }

#pragma unroll
    for (int n = 0; n < 8; ++n) {
        int col = n * 16 + l;
        float bias = b1[col];
#pragma unroll
        for (int j = 0; j < 8; ++j) {
            int row = tile * 16 + j + 8 * half;   // D layout: M = j + 8*(lane>=16)
            if (row < nnodes) {
                float v = acc[n][j] + bias;
                h[(size_t)row * HID + col] = v > 0.0f ? v : 0.0f;
            }
        }
    }
}

// ---------------- encoder head: log_phi = log_softmax((h@W2+b2)/temp) ----------------
__global__ void k_enc_head(const float* __restrict__ h, const float* __restrict__ w2,
                           const float* __restrict__ b2, float* __restrict__ log_phi,
                           int nnodes) {
    int n = blockIdx.x * blockDim.x + threadIdx.x;
    if (n >= nnodes) return;
    const float* hr = h + (size_t)n * HID;
    float lg[NC];
#pragma unroll
    for (int c = 0; c < NC; ++c) lg[c] = b2[c];
    for (int k = 0; k < HID; ++k) {
        float hv = hr[k];
#pragma unroll
        for (int c = 0; c < NC; ++c) lg[c] += hv * w2[k * NC + c];
    }
    float mx = -3.4e38f;
#pragma unroll
    for (int c = 0; c < NC; ++c) { lg[c] *= (1.0f / UNARY_TEMP_); mx = fmaxf(mx, lg[c]); }
    float s = 0.0f;
#pragma unroll
    for (int c = 0; c < NC; ++c) s += __expf(lg[c] - mx);
    float lse = mx + __logf(s);
#pragma unroll
    for (int c = 0; c < NC; ++c) log_phi[(size_t)n * NC + c] = lg[c] - lse;
}

// ---------------- degree ----------------
__global__ void k_deg(const int* __restrict__ srcArr, float* __restrict__ deg, int E) {
    int e = blockIdx.x * blockDim.x + threadIdx.x;
    if (e < E) atomicAdd(&deg[srcArr[e]], 1.0f);
}
__global__ void k_logdeg(const float* __restrict__ deg, float* __restrict__ logdeg, int n) {
    int i = blockIdx.x * blockDim.x + threadIdx.x;
    if (i < n) logdeg[i] = __logf(deg[i] + 1.0f);
}

// ---------------- edge MLP: wh = W_MAX*sigmoid(MLP([hs*hd,|hs-hd|,struct])) ----------------
__global__ void __launch_bounds__(256)
k_edge_mlp(const float* __restrict__ h, const int* __restrict__ src,
           const int* __restrict__ dst, const float* __restrict__ logdeg,
           const float* __restrict__ w1, const float* __restrict__ b1,
           const float* __restrict__ w2, const float* __restrict__ b2,
           float* __restrict__ wh, int E) {
    __shared__ float sW[EMIN * EH];   // 66048 B (CDNA5 LDS: up to 320KB/WG)
    __shared__ float sB[EH];
    __shared__ float sW2[EH];
    __shared__ float sB2;
    for (int i = threadIdx.x; i < EMIN * EH; i += blockDim.x) sW[i] = w1[i];
    for (int i = threadIdx.x; i < EH; i += blockDim.x) { sB[i] = b1[i]; sW2[i] = w2[i]; }
    if (threadIdx.x == 0) sB2 = b2[0];
    __syncthreads();

    int e = blockIdx.x * blockDim.x + threadIdx.x;
    if (e >= E) return;
    int s = src[e], d = dst[e];
    const float4* hs4 = (const float4*)(h + (size_t)s * HID);
    const float4* hd4 = (const float4*)(h + (size_t)d * HID);

    float acc[EH];
#pragma unroll
    for (int j = 0; j < EH; ++j) acc[j] = sB[j];

    for (int kk = 0; kk < HID / 4; ++kk) {
        float4 a = hs4[kk], b = hd4[kk];
#pragma unroll
        for (int t = 0; t < 4; ++t) {
            float av = (&a.x)[t], bv = (&b.x)[t];
            float p = av * bv;
            float q = fabsf(av - bv);
            const float* wp = sW + (kk * 4 + t) * EH;          // product rows 0..127
            const float* wq = sW + (HID + kk * 4 + t) * EH;    // absdiff rows 128..255
#pragma unroll
            for (int j = 0; j < EH; ++j) acc[j] += p * wp[j] + q * wq[j];
        }
    }
    float la = logdeg[s], lb = logdeg[d];
    float f1 = la + lb, f2 = fabsf(la - lb);
    {
        const float* wp = sW + 256 * EH;
        const float* wq = sW + 257 * EH;
#pragma unroll
        for (int j = 0; j < EH; ++j) acc[j] += f1 * wp[j] + f2 * wq[j];
    }
    float wr = sB2;
#pragma unroll
    for (int j = 0; j < EH; ++j) wr += fmaxf(acc[j], 0.0f) * sW2[j];
    wh[e] = W_MAX_ / (1.0f + __expf(-wr));
}

// ---------------- symmetrize w + edge norm ----------------
__global__ void k_sym_norm(const float* __restrict__ wh, const int* __restrict__ rev,
                           const int* __restrict__ src, const int* __restrict__ dst,
                           const float* __restrict__ deg, float* __restrict__ w,
                           float* __restrict__ en, int E) {
    int e = blockIdx.x * blockDim.x + threadIdx.x;
    if (e >= E) return;
    w[e] = 0.5f * (wh[e] + wh[rev[e]]);
    float dcs = fmaxf(deg[src[e]], 1.0f);
    float dcd = fmaxf(deg[dst[e]], 1.0f);
    en[e] = rsqrtf(dcs * dcd);
}

// ---------------- R = (softplus(scale)+1e-6)*tanh(0.5*(R_raw+R_raw^T)) ----------------
__global__ void k_make_R(const float* __restrict__ R_raw, const float* __restrict__ R_scale_log,
                         float* __restrict__ R) {
    int i = threadIdx.x;
    if (i < NC * NC) {
        float sc = logf(1.0f + __expf(R_scale_log[0])) + 1e-6f;
        int r = i / NC, c = i % NC;
        R[i] = sc * tanhf(0.5f * (R_raw[i] + R_raw[c * NC + r]));
    }
}

// ---------------- m0 = softmax(log_phi[src]) ----------------
__global__ void k_init_m(const float* __restrict__ log_phi, const int* __restrict__ src,
                         float* __restrict__ m, int E) {
    int e = blockIdx.x * blockDim.x + threadIdx.x;
    if (e >= E) return;
    int s = src[e];
    float lp[NC], mx = -3.4e38f;
#pragma unroll
    for (int c = 0; c < NC; ++c) { lp[c] = log_phi[(size_t)s * NC + c]; mx = fmaxf(mx, lp[c]); }
    float sum = 0.0f;
#pragma unroll
    for (int c = 0; c < NC; ++c) { lp[c] = __expf(lp[c] - mx); sum += lp[c]; }
    float inv = 1.0f / sum;
#pragma unroll
    for (int c = 0; c < NC; ++c) m[(size_t)e * NC + c] = lp[c] * inv;
}

// ---------------- BP: f = m@exp(clip(w*R)); log_f; scatter-add into sum_in ----------------
__global__ void __launch_bounds__(256)
k_edge_msg(const float* __restrict__ m, const float* __restrict__ w,
           const float* __restrict__ en, const float* __restrict__ R,
           const int* __restrict__ dst, float* __restrict__ log_f,
           float* __restrict__ sum_in, int E) {
    __shared__ float sR[NC * NC];
    for (int i = threadIdx.x; i < NC * NC; i += blockDim.x) sR[i] = R[i];
    __syncthreads();
    int e = blockIdx.x * blockDim.x + threadIdx.x;
    if (e >= E) return;
    float we = w[e], ene = en[e];
    int de = dst[e];
    float mv[NC];
#pragma unroll
    for (int c = 0; c < NC; ++c) mv[c] = m[(size_t)e * NC + c];
#pragma unroll
    for (int d = 0; d < NC; ++d) {
        float f = 0.0f;
#pragma unroll
        for (int c = 0; c < NC; ++c) {
            float t = we * sR[c * NC + d];
            t = fminf(fmaxf(t, -EXP_CLIP_), EXP_CLIP_);
            f += mv[c] * __expf(t);
        }
        if (!isfinite(f)) f = 1.0f;                         // nan_to_num
        float lf = __logf(fmaxf(f, EPS_)) * ene;
        log_f[(size_t)e * NC + d] = lf;
        atomicAdd(&sum_in[(size_t)de * NC + d], lf);
    }
}

// ---------------- BP: m update with exclusion + damping ----------------
__global__ void k_edge_update(const float* __restrict__ log_phi, const float* __restrict__ sum_in,
                              const float* __restrict__ log_f, const int* __restrict__ src,
                              const int* __restrict__ rev, const float* __restrict__ msg_logit,
                              float* __restrict__ m, int E) {
    int e = blockIdx.x * blockDim.x + threadIdx.x;
    if (e >= E) return;
    int s = src[e], r = rev[e];
    float alpha = ALPHA_MAX_ / (1.0f + __expf(-msg_logit[0]));
    float t[NC], mx = -3.4e38f;
#pragma unroll
    for (int c = 0; c < NC; ++c) {
        float excl = sum_in[(size_t)s * NC + c] - log_f[(size_t)r * NC + c];
        t[c] = log_phi[(size_t)s * NC + c] + alpha * excl;
        mx = fmaxf(mx, t[c]);
    }
    float sum = 0.0f;
#pragma unroll
    for (int c = 0; c < NC; ++c) { t[c] = __expf(t[c] - mx); sum += t[c]; }
    float inv = 1.0f / sum;
    float s2 = 0.0f;
    float mv[NC];
#pragma unroll
    for (int c = 0; c < NC; ++c) {
        float mn = t[c] * inv;
        float v = (1.0f - ETA_) * m[(size_t)e * NC + c] + ETA_ * mn;
        v = fmaxf(v, EPS_);
        mv[c] = v; s2 += v;
    }
    float i2 = 1.0f / s2;
#pragma unroll
    for (int c = 0; c < NC; ++c) m[(size_t)e * NC + c] = mv[c] * i2;
}

// ---------------- beliefs = softmax(log_phi + alpha*sum_in) ----------------
__global__ void k_beliefs(const float* __restrict__ log_phi, const float* __restrict__ sum_in,
                          const float* __restrict__ msg_logit, float* __restrict__ out, int N) {
    int n = blockIdx.x * blockDim.x + threadIdx.x;
    if (n >= N) return;
    float alpha = ALPHA_MAX_ / (1.0f + __expf(-msg_logit[0]));
    float t[NC], mx = -3.4e38f;
#pragma unroll
    for (int c = 0; c < NC; ++c) {
        t[c] = log_phi[(size_t)n * NC + c] + alpha * sum_in[(size_t)n * NC + c];
        mx = fmaxf(mx, t[c]);
    }
    float sum = 0.0f;
#pragma unroll
    for (int c = 0; c < NC; ++c) { t[c] = __expf(t[c] - mx); sum += t[c]; }
    float inv = 1.0f / sum;
#pragma unroll
    for (int c = 0; c < NC; ++c) out[(size_t)n * NC + c] = t[c] * inv;
}

// ---------------- host-side launch ----------------
extern "C" void kernel_launch(void* const* d_in, const int* in_sizes, int n_in,
                              void* d_out, int out_size, void* d_ws, size_t ws_size,
                              hipStream_t stream) {
    const float* x        = (const float*)d_in[0];
    const int*   ei       = (const int*)d_in[1];
    const int*   rev      = (const int*)d_in[2];
    const float* enc_w1   = (const float*)d_in[3];
    const float* enc_b1   = (const float*)d_in[4];
    const float* enc_w2   = (const float*)d_in[5];
    const float* enc_b2   = (const float*)d_in[6];
    const float* em_w1    = (const float*)d_in[7];
    const float* em_b1    = (const float*)d_in[8];
    const float* em_w2    = (const float*)d_in[9];
    const float* em_b2    = (const float*)d_in[10];
    const float* R_raw    = (const float*)d_in[11];
    const float* R_scale  = (const float*)d_in[12];
    const float* msg_logit= (const float*)d_in[13];

    const int E = in_sizes[2];                 // directed edges = 800000
    const int N = out_size / NC;               // 50000
    const int* src = ei;
    const int* dst = ei + E;

    // workspace layout (floats)
    float* ws      = (float*)d_ws;
    float* h       = ws;                         size_t off = (size_t)N * HID;
    float* log_phi = ws + off;                   off += (size_t)N * NC;
    float* deg     = ws + off;                   off += (size_t)N;
    float* logdeg  = ws + off;                   off += (size_t)N;
    float* wh      = ws + off;                   off += (size_t)E;
    float* w       = ws + off;                   off += (size_t)E;
    float* en      = ws + off;                   off += (size_t)E;
    float* R       = ws + off;                   off += 128;
    float* m       = ws + off;                   off += (size_t)E * NC;
    float* log_f   = ws + off;                   off += (size_t)E * NC;
    float* sum_in  = ws + off;                   off += (size_t)N * NC;
    (void)ws_size; (void)n_in;

    const int B = 256;
    const int gE  = (E + B - 1) / B;
    const int gN  = (N + B - 1) / B;
    const int gNC = (N * NC + B - 1) / B;
    const int ntiles = (N + 15) / 16;
    const int gTile  = (ntiles + 3) / 4;

    // encoder
    k_enc_gemm<<<gTile, 128, 0, stream>>>(x, enc_w1, enc_b1, h, ntiles, N);
    k_enc_head<<<gN, B, 0, stream>>>(h, enc_w2, enc_b2, log_phi, N);

    // degrees
    k_zero<<<gN, B, 0, stream>>>(deg, N);
    k_deg<<<gE, B, 0, stream>>>(src, deg, E);
    k_logdeg<<<gN, B, 0, stream>>>(deg, logdeg, N);

    // pairwise kernel weights
    k_edge_mlp<<<gE, B, 0, stream>>>(h, src, dst, logdeg, em_w1, em_b1, em_w2, em_b2, wh, E);
    k_sym_norm<<<gE, B, 0, stream>>>(wh, rev, src, dst, deg, w, en, E);
    k_make_R<<<1, 128, 0, stream>>>(R_raw, R_scale, R);

    // BP init + iterations
    k_init_m<<<gE, B, 0, stream>>>(log_phi, src, m, E);
    for (int it = 0; it < T_ITERS; ++it) {
        k_zero<<<gNC, B, 0, stream>>>(sum_in, N * NC);
        k_edge_msg<<<gE, B, 0, stream>>>(m, w, en, R, dst, log_f, sum_in, E);
        k_edge_update<<<gE, B, 0, stream>>>(log_phi, sum_in, log_f, src, rev, msg_logit, m, E);
    }

    // final beliefs
    k_zero<<<gNC, B, 0, stream>>>(sum_in, N * NC);
    k_edge_msg<<<gE, B, 0, stream>>>(m, w, en, R, dst, log_f, sum_in, E);
    k_beliefs<<<gN, B, 0, stream>>>(log_phi, sum_in, msg_logit, (float*)d_out, N);
}